// GlobalAttention_26946624815386
// MI455X (gfx1250) — compile-verified
//
#include <hip/hip_runtime.h>
#include <math.h>

// ---------------- problem constants ----------------
#define HD        512          // H
#define NPOS      65536        // 256*256 positions
#define ROWS_PB   128          // rows (positions) per block in alpha pass: 8 waves * 16
#define KC        64           // K-chunk staged in LDS per iteration

// ---------------- workspace layout (float offsets into d_ws) ----------------
#define WS_HID    0            // hid_term scalar
#define WS_GMAX   1            // global max of alpha
#define WS_INV    2            // 1 / sum(exp(alpha - gmax))
#define WS_PMAX   16           // 64 per-block max partials
#define WS_PSUM   128          // 64 per-block sum partials
#define WS_ATTN   256          // 512 floats, UNNORMALIZED attn accumulator
#define WS_ALPHA  1024         // 65536 floats: alpha, then in-place exp(alpha-gmax)

typedef __attribute__((ext_vector_type(2))) float v2f;
typedef __attribute__((ext_vector_type(8))) float v8f;
typedef __attribute__((ext_vector_type(4))) int   v4i;

// ------- optional CDNA5 async global->LDS path (guarded; ds_store fallback) -------
#if __has_builtin(__builtin_amdgcn_global_load_async_to_lds_b128) && \
    __has_builtin(__builtin_amdgcn_s_wait_asynccnt)
#define USE_ASYNC_LDS 1
typedef __attribute__((address_space(1))) v4i* gv4ip;   // global (AS1) int4*
typedef __attribute__((address_space(3))) v4i* lv4ip;   // LDS (AS3) int4*
__device__ __forceinline__ void async_cp16(const void* g, void* l) {
  // int-cast route: generic LDS pointer low 32 bits == LDS byte offset (aperture map)
  __builtin_amdgcn_global_load_async_to_lds_b128(
      (gv4ip)(unsigned long long)g,
      (lv4ip)(unsigned)(unsigned long long)l, 0, 0);
}
#else
#define USE_ASYNC_LDS 0
#endif

// ---------------------------------------------------------------------------
// Kernel 1: hid_term = dot(hid, att_w[0:512]) + att_b ; zero attn accumulator
// ---------------------------------------------------------------------------
__global__ void k_pre(const float* __restrict__ hid, const float* __restrict__ att_w,
                      const float* __restrict__ att_b, float* __restrict__ ws) {
  __shared__ float red[512];
  const int t = threadIdx.x;
  red[t] = hid[t] * att_w[t];
  __syncthreads();
  for (int s = 256; s > 0; s >>= 1) {
    if (t < s) red[t] += red[t + s];
    __syncthreads();
  }
  if (t == 0) ws[WS_HID] = red[0] + att_b[0];
  ws[WS_ATTN + t] = 0.0f;                 // zero 512-float attn accumulator
}

// ---------------------------------------------------------------------------
// Kernel 2: alpha[p] = dot(V[p,:], w_v) + hid_term  via V_WMMA_F32_16X16X4_F32
//   grid 512 x 256 threads; each wave owns 16 rows (M), B = w_v chunk broadcast
//   across all 16 columns -> dots land in column N=0 of the C/D layout.
//   V tile (128 x 64 f32, 34KB LDS, pad to stride 68 -> conflict-free b64 reads)
// ---------------------------------------------------------------------------
__global__ void __launch_bounds__(256)
k_alpha(const float* __restrict__ V, const float* __restrict__ att_w,
        float* __restrict__ ws) {
  __shared__ float tile[ROWS_PB][KC + 4];
  const float* __restrict__ wv = att_w + HD;       // w_v = att_w[0, 512:1024]
  const int t       = threadIdx.x;
  const int lane    = t & 31;
  const int wave    = t >> 5;                      // 0..7
  const int rowBase = blockIdx.x * ROWS_PB;
  const int off     = (lane >> 4) << 1;            // A layout: lanes16-31 hold K=2,3
  const int mrow    = (wave << 4) + (lane & 15);   // this lane's A-matrix row in tile
  const float hid_term = ws[WS_HID];

  v8f acc = {};
  for (int kc = 0; kc < HD; kc += KC) {
    __syncthreads();                               // previous tile fully consumed
#pragma unroll
    for (int i = 0; i < 8; ++i) {                  // 2048 float4 slots, coalesced
      const int idx = t + i * 256;
      const int r   = idx >> 4;                    // 0..127
      const int c   = (idx & 15) << 2;             // 0..60
      const float* gp = V + (size_t)(rowBase + r) * HD + kc + c;
#if USE_ASYNC_LDS
      async_cp16(gp, &tile[r][c]);                 // global_load_async_to_lds_b128
#else
      *(float4*)&tile[r][c] = *(const float4*)gp;  // vmem b128 + ds_store_b128
#endif
    }
#if USE_ASYNC_LDS
    __builtin_amdgcn_s_wait_asynccnt(0);
#endif
    __syncthreads();
#pragma unroll
    for (int k = 0; k < KC; k += 4) {              // 16 WMMAs per chunk
      v2f a = *(const v2f*)&tile[mrow][k + off];   // ds_load_b64, A fragment
      v2f b;                                        // B[k][n] = w_v[k] (all n equal)
      b.x = wv[kc + k + off];
      b.y = wv[kc + k + off + 1];
      acc = __builtin_amdgcn_wmma_f32_16x16x4_f32(
          false, a, false, b, (short)0, acc, false, false);
    }
  }
  // column N=0: lane0 -> M=0..7 in acc[0..7], lane16 -> M=8..15
  if ((lane & 15) == 0) {
    const int row = rowBase + (wave << 4) + ((lane >> 4) << 3);
#pragma unroll
    for (int i = 0; i < 8; ++i) ws[WS_ALPHA + row + i] = acc[i] + hid_term;
  }
}

// ---------------------------------------------------------------------------
// Kernels 3-6: two-level softmax normalization (kernel boundaries = global sync)
// ---------------------------------------------------------------------------
__global__ void k_max1(float* __restrict__ ws) {
  __shared__ float red[256];
  const int t = threadIdx.x;
  const float* a = ws + WS_ALPHA + blockIdx.x * 1024;
  float m = -3.0e38f;
  for (int i = t; i < 1024; i += 256) m = fmaxf(m, a[i]);
  red[t] = m; __syncthreads();
  for (int s = 128; s > 0; s >>= 1) { if (t < s) red[t] = fmaxf(red[t], red[t + s]); __syncthreads(); }
  if (t == 0) ws[WS_PMAX + blockIdx.x] = red[0];
}

__global__ void k_max2(float* __restrict__ ws) {
  __shared__ float red[64];
  const int t = threadIdx.x;
  red[t] = ws[WS_PMAX + t]; __syncthreads();
  for (int s = 32; s > 0; s >>= 1) { if (t < s) red[t] = fmaxf(red[t], red[t + s]); __syncthreads(); }
  if (t == 0) ws[WS_GMAX] = red[0];
}

__global__ void k_exp(float* __restrict__ ws) {   // in-place alpha -> exp(alpha-gmax)
  __shared__ float red[256];
  const int t = threadIdx.x;
  float* a = ws + WS_ALPHA + blockIdx.x * 1024;
  const float gmax = ws[WS_GMAX];
  float s = 0.f;
  for (int i = t; i < 1024; i += 256) { float e = __expf(a[i] - gmax); a[i] = e; s += e; }
  red[t] = s; __syncthreads();
  for (int st = 128; st > 0; st >>= 1) { if (t < st) red[t] += red[t + st]; __syncthreads(); }
  if (t == 0) ws[WS_PSUM + blockIdx.x] = red[0];
}

__global__ void k_sum2(float* __restrict__ ws) {
  __shared__ float red[64];
  const int t = threadIdx.x;
  red[t] = ws[WS_PSUM + t]; __syncthreads();
  for (int s = 32; s > 0; s >>= 1) { if (t < s) red[t] += red[t + s]; __syncthreads(); }
  if (t == 0) ws[WS_INV] = 1.0f / red[0];
}

// ---------------------------------------------------------------------------
// Kernel 7: attn += exp_weight * V rows. Second pass over V (L2-resident).
//   128 blocks x 512 rows; each lane owns 2 columns -> float2 loads cover a
//   full 2KB row per block per iteration (perfectly coalesced).
// ---------------------------------------------------------------------------
__global__ void __launch_bounds__(256)
k_attn(const float* __restrict__ V, float* __restrict__ ws) {
  const int t = threadIdx.x;
  const size_t rowBase = (size_t)blockIdx.x * 512;
  const float* __restrict__ e = ws + WS_ALPHA + rowBase;   // exp weights
  const float* __restrict__ vp = V + rowBase * HD + 2 * t;
  float2 acc = {0.f, 0.f};
#pragma unroll 4
  for (int r = 0; r < 512; ++r) {
    if ((r & 63) == 0) __builtin_prefetch(vp + (size_t)64 * HD, 0, 3); // global_prefetch_b8
    const float w = e[r];
    const float2 v = *(const float2*)vp;
    acc.x = fmaf(w, v.x, acc.x);
    acc.y = fmaf(w, v.y, acc.y);
    vp += HD;
  }
  atomicAdd(&ws[WS_ATTN + 2 * t],     acc.x);   // global_atomic_add_f32
  atomicAdd(&ws[WS_ATTN + 2 * t + 1], acc.y);
}

// ---------------------------------------------------------------------------
// Kernel 8: ctx[o] = relu( sum_j cat[j]*ctx_w[o,j] + ctx_b[o] ), WMMA again.
//   cat = [inp | attn/sum]; 4 blocks x 8 waves x 16 output rows.
// ---------------------------------------------------------------------------
__global__ void __launch_bounds__(256)
k_ctx(const float* __restrict__ inp, const float* __restrict__ ctx_w,
      const float* __restrict__ ctx_b, const float* __restrict__ ws,
      float* __restrict__ out) {
  const int t     = threadIdx.x;
  const int lane  = t & 31;
  const int wave  = t >> 5;
  const int oBase = blockIdx.x * 128 + (wave << 4);
  const int off   = (lane >> 4) << 1;
  const int orow  = oBase + (lane & 15);
  const float* __restrict__ wrow = ctx_w + (size_t)orow * (2 * HD);
  const float* __restrict__ attn = ws + WS_ATTN;
  const float inv = ws[WS_INV];

  v8f acc = {};
  for (int k = 0; k < 2 * HD; k += 4) {
    v2f a = *(const v2f*)(wrow + k + off);
    const int j = k + off;
    v2f b;
    if (j < HD) { b.x = inp[j];             b.y = inp[j + 1]; }
    else        { b.x = attn[j - HD] * inv; b.y = attn[j - HD + 1] * inv; }
    acc = __builtin_amdgcn_wmma_f32_16x16x4_f32(
        false, a, false, b, (short)0, acc, false, false);
  }
  if ((lane & 15) == 0) {
    const int o = oBase + ((lane >> 4) << 3);
#pragma unroll
    for (int i = 0; i < 8; ++i) out[o + i] = fmaxf(acc[i] + ctx_b[o + i], 0.0f);
  }
}

// ---------------------------------------------------------------------------
extern "C" void kernel_launch(void* const* d_in, const int* in_sizes, int n_in,
                              void* d_out, int out_size, void* d_ws, size_t ws_size,
                              hipStream_t stream) {
  (void)in_sizes; (void)n_in; (void)out_size; (void)ws_size;
  const float* inp   = (const float*)d_in[0];
  const float* hid   = (const float*)d_in[1];
  const float* V     = (const float*)d_in[2];
  const float* att_w = (const float*)d_in[3];
  const float* att_b = (const float*)d_in[4];
  const float* ctx_w = (const float*)d_in[5];
  const float* ctx_b = (const float*)d_in[6];
  float* ws  = (float*)d_ws;
  float* out = (float*)d_out;

  k_pre  <<<1,   512, 0, stream>>>(hid, att_w, att_b, ws);
  k_alpha<<<NPOS / ROWS_PB, 256, 0, stream>>>(V, att_w, ws);   // 512 blocks
  k_max1 <<<64,  256, 0, stream>>>(ws);
  k_max2 <<<1,    64, 0, stream>>>(ws);
  k_exp  <<<64,  256, 0, stream>>>(ws);
  k_sum2 <<<1,    64, 0, stream>>>(ws);
  k_attn <<<128, 256, 0, stream>>>(V, ws);
  k_ctx  <<<4,   256, 0, stream>>>(inp, ctx_w, ctx_b, ws, out);
}